// SingleFixationLoss_91319594648376
// MI455X (gfx1250) — compile-verified
//
#include <hip/hip_runtime.h>
#include <cstdint>

typedef float v2f __attribute__((ext_vector_type(2)));
typedef float v8f __attribute__((ext_vector_type(8)));

#define NSLICE 640     // B*T
#define HW     4096    // H*W
#define EPSF   1e-8f

__device__ __forceinline__ float wave_sum(float v) {
  #pragma unroll
  for (int o = 16; o > 0; o >>= 1) v += __shfl_down(v, o, 32);
  return v;
}
__device__ __forceinline__ float wave_max(float v) {
  #pragma unroll
  for (int o = 16; o > 0; o >>= 1) v = fmaxf(v, __shfl_down(v, o, 32));
  return v;
}

// One block per (b,t) slice. 256 threads, each owns 16 contiguous elements.
__global__ __launch_bounds__(256) void slice_kernel(const float* __restrict__ pred,
                                                    const float* __restrict__ tgt,
                                                    float* __restrict__ ws) {
  __shared__ float s_tgt[HW];        // 16 KB async-staged target slice
  __shared__ float s_red[8][10];
  __shared__ float s_bcast;

  const int t     = threadIdx.x;
  const int lane  = t & 31;
  const int wv    = t >> 5;
  const int slice = blockIdx.x;
  const size_t base = (size_t)slice * HW;
  const int e0 = t * 16;

  // ---- issue async global->LDS copy of the target slice (ASYNCcnt path).
  // Each thread copies its own 64 bytes; it later reads back only bytes its
  // own wave's async instructions wrote, so s_wait_asynccnt 0 is sufficient.
  {
    uint32_t lds0 = (uint32_t)(uintptr_t)(&s_tgt[0]) + (uint32_t)(t * 64);
    uint64_t g0   = (uint64_t)(uintptr_t)(tgt + base + e0);
    #pragma unroll
    for (int r = 0; r < 4; ++r) {
      uint32_t la = lds0 + (uint32_t)(r * 16);
      uint64_t ga = g0 + (uint64_t)(r * 16);
      asm volatile("global_load_async_to_lds_b128 %0, %1, off"
                   :: "v"(la), "v"(ga) : "memory");
    }
  }

  // ---- phase 1 (overlapped with async copy): sigmoid + p-only accumulators
  float pr[16], pp[16];
  float sum_p = 0.f, max_p = 0.f, pxa = 0.f, pya = 0.f;
  #pragma unroll
  for (int r = 0; r < 4; ++r) {
    float4 v = *(const float4*)(pred + base + e0 + r * 4);
    float xs[4] = {v.x, v.y, v.z, v.w};
    #pragma unroll
    for (int j = 0; j < 4; ++j) {
      int   i   = r * 4 + j;
      int   idx = e0 + i;
      float x   = xs[j];
      float p   = 1.0f / (1.0f + __expf(-x));
      pr[i] = x; pp[i] = p;
      sum_p += p;
      max_p  = fmaxf(max_p, p);
      pxa   += p * (float)(idx & 63);   // w coordinate
      pya   += p * (float)(idx >> 6);   // h coordinate
    }
  }

  asm volatile("s_wait_asynccnt 0x0" ::: "memory");
  __syncthreads();

  // ---- phase 2: target-dependent terms from LDS (pred/p still in registers)
  float focal = 0.f, bg = 0.f, sum_t = 0.f, txa = 0.f, tya = 0.f;
  const float4* st4 = (const float4*)s_tgt;
  #pragma unroll
  for (int r = 0; r < 4; ++r) {
    float4 tv = st4[(e0 >> 2) + r];
    float ts[4] = {tv.x, tv.y, tv.z, tv.w};
    #pragma unroll
    for (int j = 0; j < 4; ++j) {
      int   i   = r * 4 + j;
      int   idx = e0 + i;
      float x = pr[i], p = pp[i], tg = ts[j];
      float bce = fmaxf(x, 0.f) - x * tg + log1pf(__expf(-fabsf(x)));
      float pt  = (tg == 1.f) ? p : (1.f - p);
      float om  = 1.f - pt;
      focal += om * om * bce;                 // alpha=1, gamma=2
      float pb = (tg == 0.f) ? p : 0.f;
      bg    += pb * pb;
      sum_t += tg;
      txa   += tg * (float)(idx & 63);
      tya   += tg * (float)(idx >> 6);
    }
  }

  // ---- block reduction: 8 sums + 1 max (wave32 shuffle, then 8-wave LDS)
  float a0 = wave_sum(focal), a1 = wave_sum(bg),   a2 = wave_sum(sum_p);
  float a3 = wave_sum(pxa),   a4 = wave_sum(pya),  a5 = wave_sum(sum_t);
  float a6 = wave_sum(txa),   a7 = wave_sum(tya),  a8 = wave_max(max_p);
  if (lane == 0) {
    s_red[wv][0]=a0; s_red[wv][1]=a1; s_red[wv][2]=a2; s_red[wv][3]=a3;
    s_red[wv][4]=a4; s_red[wv][5]=a5; s_red[wv][6]=a6; s_red[wv][7]=a7;
    s_red[wv][8]=a8;
  }
  __syncthreads();
  if (t == 0) {
    float F=0,Bk=0,Sp=0,Px=0,Py=0,St=0,Tx=0,Ty=0,Mx=0;
    for (int w = 0; w < 8; ++w) {
      F+=s_red[w][0]; Bk+=s_red[w][1]; Sp+=s_red[w][2]; Px+=s_red[w][3];
      Py+=s_red[w][4]; St+=s_red[w][5]; Tx+=s_red[w][6]; Ty+=s_red[w][7];
      Mx = fmaxf(Mx, s_red[w][8]);
    }
    float SpE = Sp + EPSF, StE = St + EPSF;
    float dx = Px / SpE - Tx / StE;
    float dy = Py / SpE - Ty / StE;
    ws[0*NSLICE + slice] = F;                      // focal partial sum
    ws[1*NSLICE + slice] = Bk;                     // background partial sum
    ws[3*NSLICE + slice] = 1.f - Mx;               // concentration (done)
    ws[4*NSLICE + slice] = sqrtf(dx*dx + dy*dy);   // coordinate (done)
    s_bcast = 1.f / SpE;
  }
  __syncthreads();

  // ---- entropy pass: needs the slice sum, p values still in registers
  float invS = s_bcast;
  float ent = 0.f;
  #pragma unroll
  for (int i = 0; i < 16; ++i) {
    float pn = pp[i] * invS;
    ent -= pn * __logf(pn + EPSF);
  }
  float e = wave_sum(ent);
  if (lane == 0) s_red[wv][0] = e;
  __syncthreads();
  if (t == 0) {
    float E = 0.f;
    for (int w = 0; w < 8; ++w) E += s_red[w][0];
    ws[2*NSLICE + slice] = E;                      // entropy (done)
  }
}

// One wave32. Reduce 5 arrays of 640 via chained V_WMMA_F32_16X16X4_F32 with
// an all-ones A: D[m,n] = sum_k B[k,n] + C[m,n]; the B lane->(k,n) mapping is
// irrelevant for a total sum, so 64 consecutive values per step are summed
// exactly once. EXEC is all-ones (32 threads, no divergence around WMMA).
__global__ __launch_bounds__(32) void finalize_kernel(const float* __restrict__ ws,
                                                      float* __restrict__ out) {
  const int lane = threadIdx.x;
  v2f ones; ones[0] = 1.f; ones[1] = 1.f;
  float sums[5];
  #pragma unroll
  for (int q = 0; q < 5; ++q) {
    const float* a = ws + q * NSLICE;
    v8f acc = {};
    #pragma unroll
    for (int c = 0; c < 10; ++c) {        // 10 * 64 = 640 values
      v2f b;
      b[0] = a[c * 64 + lane];
      b[1] = a[c * 64 + 32 + lane];
      acc = __builtin_amdgcn_wmma_f32_16x16x4_f32(
          false, ones, false, b, (short)0, acc, false, false);
    }
    float v = acc[0];                     // row M=0: D[0,n] in lanes 0..15
    v += __shfl_xor(v, 1, 16);
    v += __shfl_xor(v, 2, 16);
    v += __shfl_xor(v, 4, 16);
    v += __shfl_xor(v, 8, 16);
    sums[q] = v;                          // lane 0 holds the total
  }
  if (lane == 0) {
    const float invN = 1.f / (float)(NSLICE * HW);
    const float invS = 1.f / (float)NSLICE;
    float total = sums[0] * invN               // focal mean
                + 0.1f * sums[1] * invN        // BG_W * background mean
                + 0.8f * sums[2] * invS        // SPARSITY_W * entropy mean
                + 1.5f * sums[3] * invS        // CONC_W * concentration mean
                + 1.0f * sums[4] * invS;       // COORD_W * coordinate mean
    out[0] = total;
  }
}

extern "C" void kernel_launch(void* const* d_in, const int* in_sizes, int n_in,
                              void* d_out, int out_size, void* d_ws, size_t ws_size,
                              hipStream_t stream) {
  const float* pred = (const float*)d_in[0];
  const float* tgt  = (const float*)d_in[1];
  float* ws  = (float*)d_ws;     // needs 5*640*4 = 12.8 KB
  float* out = (float*)d_out;
  slice_kernel<<<NSLICE, 256, 0, stream>>>(pred, tgt, ws);
  finalize_kernel<<<1, 32, 0, stream>>>(ws, out);
}